// IOU3DLoss_19215683682794
// MI455X (gfx1250) — compile-verified
//
#include <hip/hip_runtime.h>
#include <math.h>

#define BLOCK 128
#define TILE  (BLOCK * 7)          // floats per input per block
#define NVERT 24                   // 8 corners + 16 edge intersections max
#define EPSF  1e-8f

// ---- gfx1250 async global->LDS staging (codegen-confirmed: global_load_async_to_lds_b32) ----
#if defined(__gfx1250__) && __has_builtin(__builtin_amdgcn_global_load_async_to_lds_b32)
#define USE_ASYNC_LDS 1
typedef __attribute__((address_space(1))) int GASI;   // global (AS1) word pointer
typedef __attribute__((address_space(3))) int LASI;   // LDS    (AS3) word pointer
#define GPTR(p) ((GASI*)(p))
#define LPTR(p) ((LASI*)(p))
#else
#define USE_ASYNC_LDS 0
#endif

// Monotone-in-angle key, same ordering as atan2(y,x) over (-pi, pi]:
// range [-2, 2]; avoids a ~30-op atan2 sequence per vertex.
__device__ __forceinline__ float pseudo_angle(float x, float y) {
    float d = fabsf(x) + fabsf(y);
    float r = x / d;
    float k = copysignf(1.0f - r, y);
    return (d == 0.0f) ? 0.0f : k;       // atan2(0,0) == 0 in the reference
}

__device__ __forceinline__ void make_corners(float x, float y, float w, float h,
                                             float alpha, float* cx, float* cy) {
    float s, c;
    __sincosf(alpha, &s, &c);
    const float hw = 0.5f * w, hh = 0.5f * h;
    const float xs[4] = { hw, -hw, -hw,  hw};
    const float ys[4] = { hh,  hh, -hh, -hh};
#pragma unroll
    for (int i = 0; i < 4; ++i) {
        cx[i] = xs[i] * c - ys[i] * s + x;
        cy[i] = xs[i] * s + ys[i] * c + y;
    }
}

// which corners of box P lie inside box Q (corners a=q0, b=q1, d=q3)
__device__ __forceinline__ void corners_in(const float* px, const float* py,
                                           const float* qx, const float* qy,
                                           bool* in) {
    const float ax = qx[0], ay = qy[0];
    const float abx = qx[1] - ax, aby = qy[1] - ay;
    const float adx = qx[3] - ax, ady = qy[3] - ay;
    const float iab = 1.0f / (abx * abx + aby * aby);
    const float iad = 1.0f / (adx * adx + ady * ady);
    const float tol = 1e-6f;
#pragma unroll
    for (int k = 0; k < 4; ++k) {
        float amx = px[k] - ax, amy = py[k] - ay;
        float pab = (abx * amx + aby * amy) * iab;
        float pad = (adx * amx + ady * amy) * iad;
        in[k] = (pab > -tol) & (pab < 1.0f + tol) & (pad > -tol) & (pad < 1.0f + tol);
    }
}

__global__ __launch_bounds__(BLOCK) void iou3d_partial_kernel(
    const float* __restrict__ pred, const float* __restrict__ targ,
    float* __restrict__ partial, int n) {
    __shared__ float sP[TILE];
    __shared__ float sT[TILE];
    // per-thread dynamic vertex scratch lives in LDS (scratch mem would hit HBM)
    __shared__ float vxs[NVERT * BLOCK];
    __shared__ float vys[NVERT * BLOCK];
    __shared__ float vks[NVERT * BLOCK];
    __shared__ float red[BLOCK];

    const int tid   = threadIdx.x;
    const int base  = blockIdx.x * TILE;   // float offset of this block's tile
    const int total = n * 7;

    // ---- coalesced staging of the 7-float rows via CDNA5 async global->LDS ----
#if USE_ASYNC_LDS
#pragma unroll
    for (int k = 0; k < 7; ++k) {
        int idx = k * BLOCK + tid;
        int g = base + idx;
        if (g < total) {
            __builtin_amdgcn_global_load_async_to_lds_b32(
                GPTR(pred + g), LPTR(&sP[idx]), 0, 0);
            __builtin_amdgcn_global_load_async_to_lds_b32(
                GPTR(targ + g), LPTR(&sT[idx]), 0, 0);
        }
    }
#if __has_builtin(__builtin_amdgcn_s_wait_asynccnt)
    __builtin_amdgcn_s_wait_asynccnt(0);
#else
    asm volatile("s_wait_asynccnt 0" ::: "memory");
#endif
#else
#pragma unroll
    for (int k = 0; k < 7; ++k) {
        int idx = k * BLOCK + tid;
        int g = base + idx;
        if (g < total) { sP[idx] = pred[g]; sT[idx] = targ[g]; }
    }
#endif
    __syncthreads();

    const int row = blockIdx.x * BLOCK + tid;
    float loss = 0.0f;

    if (row < n) {
        const float* b1 = &sP[tid * 7];
        const float* b2 = &sT[tid * 7];

        // 2D boxes: x,y,w,h,alpha = b[0],b[1],b[3],b[4],b[6]
        float c1x[4], c1y[4], c2x[4], c2y[4];
        make_corners(b1[0], b1[1], b1[3], b1[4], b1[6], c1x, c1y);
        make_corners(b2[0], b2[1], b2[3], b2[4], b2[6], c2x, c2y);

#define VX(k) vxs[(k) * BLOCK + tid]
#define VY(k) vys[(k) * BLOCK + tid]
#define VK(k) vks[(k) * BLOCK + tid]

        int   m  = 0;
        float sx = 0.0f, sy = 0.0f;

        // Branchless push: always write to the next-free slot; advance only when
        // selected. Garbage in slot m is overwritten by the next valid push and
        // slots >= final m are never read. Avoids exec-mask save/restore per
        // candidate under wave32.
#define PUSH(PXV, PYV, SEL)                                     \
        {                                                       \
            float _px = (PXV), _py = (PYV);                     \
            bool  _s  = (SEL);                                  \
            VX(m) = _px; VY(m) = _py;                           \
            sx += _s ? _px : 0.0f;                              \
            sy += _s ? _py : 0.0f;                              \
            m  += (int)_s;                                      \
        }

        bool in1[4], in2[4];
        corners_in(c1x, c1y, c2x, c2y, in1);
        corners_in(c2x, c2y, c1x, c1y, in2);
#pragma unroll
        for (int k = 0; k < 4; ++k) PUSH(c1x[k], c1y[k], in1[k]);
#pragma unroll
        for (int k = 0; k < 4; ++k) PUSH(c2x[k], c2y[k], in2[k]);

        // 4x4 edge-pair intersections (formulas match the reference exactly).
        // num==0 needs no branch: t,u become Inf/NaN and the open-interval
        // compares reject them, same as the reference's t=u=-1 path.
#pragma unroll
        for (int i = 0; i < 4; ++i) {
            float x1 = c1x[i], y1 = c1y[i];
            float x2 = c1x[(i + 1) & 3], y2 = c1y[(i + 1) & 3];
            float ex = x2 - x1, ey = y2 - y1;
#pragma unroll
            for (int j = 0; j < 4; ++j) {
                float x3 = c2x[j], y3 = c2y[j];
                float x4 = c2x[(j + 1) & 3], y4 = c2y[(j + 1) & 3];
                float num   = ex * (y3 - y1) - ey * (x3 - x1);
                float den_t = ex * (y3 - y4) - ey * (x3 - x4);
                float den_u = (x4 - x3) * (y1 - y3) - (y4 - y3) * (x1 - x3);
                float t = den_t / num;
                float u = den_u / num;
                bool sel = (num != 0.0f) & (t > 0.0f) & (t < 1.0f)
                                         & (u > 0.0f) & (u < 1.0f);
                float te = den_t / (num + EPSF);
                PUSH(x1 + te * ex, y1 + te * ey, sel);
            }
        }
#undef PUSH

        float area = 0.0f;
        if (m >= 3) {
            float invm = 1.0f / (float)m;
            float mx = sx * invm, my = sy * invm;
            // center + pseudo-angle key
            for (int k = 0; k < m; ++k) {
                float x = VX(k) - mx, y = VY(k) - my;
                VX(k) = x; VY(k) = y; VK(k) = pseudo_angle(x, y);
            }
            // insertion sort by angle (m is typically <= 8)
            for (int k = 1; k < m; ++k) {
                float tx = VX(k), ty = VY(k), tk = VK(k);
                int j = k - 1;
                while (j >= 0 && VK(j) > tk) {
                    VX(j + 1) = VX(j); VY(j + 1) = VY(j); VK(j + 1) = VK(j);
                    --j;
                }
                VX(j + 1) = tx; VY(j + 1) = ty; VK(j + 1) = tk;
            }
            // shoelace around the centroid, wrapping last->first
            float cr = 0.0f;
            float px = VX(m - 1), py = VY(m - 1);
            for (int k = 0; k < m; ++k) {
                float x = VX(k), y = VY(k);
                cr += px * y - py * x;
                px = x; py = y;
            }
            area = 0.5f * fabsf(cr);
        }

        // z overlap + volumes
        float z1 = b1[2], l1 = b1[5], z2 = b2[2], l2 = b2[5];
        float zmax = fminf(z1 + 0.5f * l1, z2 + 0.5f * l2);
        float zmin = fmaxf(z1 - 0.5f * l1, z2 - 0.5f * l2);
        float inter3 = area * fmaxf(zmax - zmin, 0.0f);
        float v1 = b1[3] * b1[4] * b1[5];
        float v2 = b2[3] * b2[4] * b2[5];
        float iou = inter3 / (v1 + v2 - inter3);
        iou = isnan(iou) ? 0.0f : iou;     // jnp.nan_to_num(nan=0.0), branchless
        loss = 1.0f - iou;
    }

    // deterministic block tree-reduction
    red[tid] = loss;
    __syncthreads();
#pragma unroll
    for (int s = BLOCK / 2; s > 0; s >>= 1) {
        if (tid < s) red[tid] += red[tid + s];
        __syncthreads();
    }
    if (tid == 0) partial[blockIdx.x] = red[0];
}

__global__ __launch_bounds__(BLOCK) void reduce_mean_kernel(
    const float* __restrict__ partial, int nparts, float* __restrict__ out, float invN) {
    __shared__ float red[BLOCK];
    float s = 0.0f;
    for (int i = threadIdx.x; i < nparts; i += BLOCK) s += partial[i];
    red[threadIdx.x] = s;
    __syncthreads();
#pragma unroll
    for (int k = BLOCK / 2; k > 0; k >>= 1) {
        if (threadIdx.x < k) red[threadIdx.x] += red[threadIdx.x + k];
        __syncthreads();
    }
    if (threadIdx.x == 0) out[0] = red[0] * invN;
}

extern "C" void kernel_launch(void* const* d_in, const int* in_sizes, int n_in,
                              void* d_out, int out_size, void* d_ws, size_t ws_size,
                              hipStream_t stream) {
    (void)n_in; (void)out_size; (void)ws_size;
    const float* pred = (const float*)d_in[0];
    const float* targ = (const float*)d_in[1];
    float* out = (float*)d_out;
    float* partial = (float*)d_ws;                 // nblocks floats of scratch
    const int n = in_sizes[0] / 7;
    const int nblocks = (n + BLOCK - 1) / BLOCK;
    iou3d_partial_kernel<<<nblocks, BLOCK, 0, stream>>>(pred, targ, partial, n);
    reduce_mean_kernel<<<1, BLOCK, 0, stream>>>(partial, nblocks, out, 1.0f / (float)n);
}